// ColorSobelLossMean_24558622998862
// MI455X (gfx1250) — compile-verified
//
#include <hip/hip_runtime.h>
#include <cstdint>

// Problem constants from the reference.
#define BATCH 16
#define H 512
#define W 512
#define T  32   // output tile
#define GT 36   // gray tile  (halo 2 each side)
#define MT 34   // sobel/mask tile (halo 1 each side)
#define NB ((W/T)*(H/T)*BATCH)   // 16*16*16 = 4096 blocks

// ---- CDNA5 async global->LDS path (guarded; fallbacks keep compile green) ----
#if defined(__has_builtin)
#  if __has_builtin(__builtin_amdgcn_global_load_async_to_lds_b128)
#    define HAVE_ASYNC_B128 1
#  else
#    define HAVE_ASYNC_B128 0
#  endif
#  if __has_builtin(__builtin_amdgcn_global_load_async_to_lds_b32)
#    define HAVE_ASYNC_B32 1
#  else
#    define HAVE_ASYNC_B32 0
#  endif
#  if __has_builtin(__builtin_amdgcn_s_wait_asynccnt)
#    define HAVE_WAIT_ASYNC 1
#  else
#    define HAVE_WAIT_ASYNC 0
#  endif
#else
#  define HAVE_ASYNC_B128 0
#  define HAVE_ASYNC_B32 0
#  define HAVE_WAIT_ASYNC 0
#endif
#define HAVE_ASYNC_ANY (HAVE_ASYNC_B128 || HAVE_ASYNC_B32)

// Clang's b32 prototype wants AS-qualified int* ("__device__ int *" = AS1);
// by analogy the b128 form takes AS-qualified int4*.
typedef int v4i __attribute__((ext_vector_type(4)));
typedef __attribute__((address_space(1))) int  gint_t;   // global dword
typedef __attribute__((address_space(3))) int  lint_t;   // LDS dword
typedef __attribute__((address_space(1))) v4i  gv4i_t;   // global 4-dword
typedef __attribute__((address_space(3))) v4i  lv4i_t;   // LDS 4-dword

__device__ __forceinline__ float u8q(float v) {
    // torch mul(255).clamp(0,255) then trunc-to-uint8 (values >= 0 so floor==trunc)
    return floorf(fminf(fmaxf(v * 255.0f, 0.0f), 255.0f));
}
__device__ __forceinline__ int refl(int i, int n) {  // BORDER_REFLECT_101
    if (i < 0)  return -i;
    if (i >= n) return 2 * n - 2 - i;
    return i;
}

__global__ __launch_bounds__(256)
void sobel_tile_kernel(const float* __restrict__ pred,
                       const float* __restrict__ trut,
                       float* __restrict__ partial)
{
    __shared__ float gP[GT][GT + 1];     // gray (pred), halo 2   -- reused for blur-h
    __shared__ float gQ[GT][GT + 1];     // gray (true)
    __shared__ float sP[MT][MT + 1];     // sobelxy (pred), halo 1
    __shared__ float sQ[MT][MT + 1];
    __shared__ float mP[MT][MT + 1];     // threshold mask, halo 1
    __shared__ float mQ[MT][MT + 1];
    __shared__ float cP[3][T][T];        // center color pixels (async-staged, 16B rows)
    __shared__ float cQ[3][T][T];
    __shared__ float redS[256];
    __shared__ float redC[256];

    const int t  = threadIdx.x;
    const int tx = blockIdx.x, ty = blockIdx.y, b = blockIdx.z;
    const int ox = tx * T, oy = ty * T;
    const size_t imgBase = (size_t)b * 3 * H * W;
    const size_t chStride = (size_t)H * W;

    // ---- stage 0: stage center color pixels into LDS (async; overlaps stages 1-3) ----
#if HAVE_ASYNC_B128
    {
        // One 16B quad per thread per channel per image: 6 async b128 ops/thread.
        const int py = t >> 3;            // t in [0,256): 32 rows x 8 quads
        const int px = (t & 7) << 2;      // 16B-aligned column within tile
        const size_t off = imgBase + (size_t)(oy + py) * W + (ox + px);
        #pragma unroll
        for (int c = 0; c < 3; ++c) {
            const size_t o = off + (size_t)c * chStride;
            __builtin_amdgcn_global_load_async_to_lds_b128(
                (gv4i_t*)(uintptr_t)(pred + o),
                (lv4i_t*)(uint32_t)(uintptr_t)&cP[c][py][px], 0, 0);
            __builtin_amdgcn_global_load_async_to_lds_b128(
                (gv4i_t*)(uintptr_t)(trut + o),
                (lv4i_t*)(uint32_t)(uintptr_t)&cQ[c][py][px], 0, 0);
        }
    }
#elif HAVE_ASYNC_B32
    for (int p = t; p < T * T; p += 256) {
        const int py = p >> 5, px = p & 31;
        const size_t off = imgBase + (size_t)(oy + py) * W + (ox + px);
        #pragma unroll
        for (int c = 0; c < 3; ++c) {
            const size_t o = off + (size_t)c * chStride;
            __builtin_amdgcn_global_load_async_to_lds_b32(
                (gint_t*)(uintptr_t)(pred + o),
                (lint_t*)(uint32_t)(uintptr_t)&cP[c][py][px], 0, 0);
            __builtin_amdgcn_global_load_async_to_lds_b32(
                (gint_t*)(uintptr_t)(trut + o),
                (lint_t*)(uint32_t)(uintptr_t)&cQ[c][py][px], 0, 0);
        }
    }
#else
    for (int p = t; p < T * T; p += 256) {
        const int py = p >> 5, px = p & 31;
        const size_t off = imgBase + (size_t)(oy + py) * W + (ox + px);
        #pragma unroll
        for (int c = 0; c < 3; ++c) {
            const size_t o = off + (size_t)c * chStride;
            cP[c][py][px] = pred[o];
            cQ[c][py][px] = trut[o];
        }
    }
#endif

    // ---- stage 1: u8-quantized grayscale over 36x36 (halo 2, reflect-101) ----
    for (int i = t; i < GT * GT; i += 256) {
        const int yy = i / GT, xx = i % GT;
        const int gy = refl(oy + yy - 2, H);
        const int gx = refl(ox + xx - 2, W);
        const size_t o0 = imgBase + (size_t)gy * W + gx;
        float r = u8q(pred[o0]);
        float g = u8q(pred[o0 + chStride]);
        float bl = u8q(pred[o0 + 2 * chStride]);
        gP[yy][xx] = rintf(0.299f * r + 0.587f * g + 0.114f * bl);
        r  = u8q(trut[o0]);
        g  = u8q(trut[o0 + chStride]);
        bl = u8q(trut[o0 + 2 * chStride]);
        gQ[yy][xx] = rintf(0.299f * r + 0.587f * g + 0.114f * bl);
    }
    __syncthreads();

    // ---- stage 2: Sobel (ksize=1) + addWeighted + threshold over 34x34 (halo 1) ----
    for (int i = t; i < MT * MT; i += 256) {
        const int ys = i / MT, xs = i % MT;
        float sx = fminf(fabsf(gP[ys + 1][xs + 2] - gP[ys + 1][xs]), 255.0f);
        float sy = fminf(fabsf(gP[ys + 2][xs + 1] - gP[ys][xs + 1]), 255.0f);
        float so = rintf(0.5f * sx + 0.5f * sy);
        sP[ys][xs] = so;
        mP[ys][xs] = so > 10.0f ? 255.0f : 0.0f;
        sx = fminf(fabsf(gQ[ys + 1][xs + 2] - gQ[ys + 1][xs]), 255.0f);
        sy = fminf(fabsf(gQ[ys + 2][xs + 1] - gQ[ys][xs + 1]), 255.0f);
        so = rintf(0.5f * sx + 0.5f * sy);
        sQ[ys][xs] = so;
        mQ[ys][xs] = so > 10.0f ? 255.0f : 0.0f;
    }
    __syncthreads();

    // ---- stage 3a: horizontal [1,2,1]/4 blur of mask; reuse gray LDS as scratch ----
    float* bhP = &gP[0][0];   // 34 x 32, stride T (gray is dead now)
    float* bhQ = &gQ[0][0];
    for (int i = t; i < MT * T; i += 256) {
        const int y = i / T, x = i % T;
        bhP[y * T + x] = 0.25f * mP[y][x] + 0.5f * mP[y][x + 1] + 0.25f * mP[y][x + 2];
        bhQ[y * T + x] = 0.25f * mQ[y][x] + 0.5f * mQ[y][x + 1] + 0.25f * mQ[y][x + 2];
    }
#if HAVE_ASYNC_ANY
#  if HAVE_WAIT_ASYNC
    __builtin_amdgcn_s_wait_asynccnt(0);
#  else
    asm volatile("s_wait_asynccnt 0x0" ::: "memory");
#  endif
#endif
    __syncthreads();   // each wave waited its own ASYNCcnt first -> LDS coherent

    // ---- stage 3b + 4: vertical blur, keep-mask, fused loss accumulation ----
    float sumS = 0.0f, sumC = 0.0f;
    for (int p = t; p < T * T; p += 256) {
        const int y = p >> 5, x = p & 31;
        const float blurP = rintf(0.25f * bhP[y * T + x] + 0.5f * bhP[(y + 1) * T + x] +
                                  0.25f * bhP[(y + 2) * T + x]);
        const float blurQ = rintf(0.25f * bhQ[y * T + x] + 0.5f * bhQ[(y + 1) * T + x] +
                                  0.25f * bhQ[(y + 2) * T + x]);
        sumS += fabsf(sP[y + 1][x + 1] - sQ[y + 1][x + 1]);
        const float kP = blurP > 0.0f ? 1.0f : 0.0f;
        const float kQ = blurQ > 0.0f ? 1.0f : 0.0f;
        #pragma unroll
        for (int c = 0; c < 3; ++c)
            sumC += fabsf(u8q(cP[c][y][x]) * kP - u8q(cQ[c][y][x]) * kQ);
    }

    // ---- deterministic block reduction ----
    redS[t] = sumS; redC[t] = sumC;
    __syncthreads();
    for (int s = 128; s > 0; s >>= 1) {
        if (t < s) { redS[t] += redS[t + s]; redC[t] += redC[t + s]; }
        __syncthreads();
    }
    if (t == 0) {
        const int blk = (b * (int)gridDim.y + ty) * (int)gridDim.x + tx;
        partial[2 * blk]     = redS[0];
        partial[2 * blk + 1] = redC[0];
    }
}

__global__ __launch_bounds__(256)
void final_reduce_kernel(const float* __restrict__ partial, float* __restrict__ out)
{
    __shared__ float rS[256], rC[256];
    const int t = threadIdx.x;
    float sS = 0.0f, sC = 0.0f;
    for (int i = t; i < NB; i += 256) { sS += partial[2 * i]; sC += partial[2 * i + 1]; }
    rS[t] = sS; rC[t] = sC;
    __syncthreads();
    for (int s = 128; s > 0; s >>= 1) {
        if (t < s) { rS[t] += rS[t + s]; rC[t] += rC[t + s]; }
        __syncthreads();
    }
    if (t == 0) {
        const float nPix = (float)BATCH * (float)H * (float)W;
        out[0] = rS[0] / (255.0f * nPix);          // loss_sobel
        out[1] = rC[0] / (255.0f * nPix * 3.0f);   // loss_color_edge
    }
}

extern "C" void kernel_launch(void* const* d_in, const int* in_sizes, int n_in,
                              void* d_out, int out_size, void* d_ws, size_t ws_size,
                              hipStream_t stream)
{
    const float* pred = (const float*)d_in[0];
    const float* trut = (const float*)d_in[1];
    float* partial = (float*)d_ws;                 // NB*2 floats = 32 KB scratch

    dim3 grid(W / T, H / T, BATCH);                // 16 x 16 x 16 tiles
    sobel_tile_kernel<<<grid, 256, 0, stream>>>(pred, trut, partial);
    final_reduce_kernel<<<1, 256, 0, stream>>>(partial, (float*)d_out);
}